// MDMLPPatch_46059229282884
// MI455X (gfx1250) — compile-verified
//
#include <hip/hip_runtime.h>

typedef __attribute__((ext_vector_type(2))) float v2f;
typedef __attribute__((ext_vector_type(8))) float v8f;

#define C_CH  64
#define HH    56
#define WW    56
#define P_TOT (HH * WW)      // 3136 output positions per (n,c)

// One wave32 computes a 16x64 output tile of the M=(n,c,p) x N=64 GEMM with K=9
// (padded to 12) using V_WMMA_F32_16X16X4_F32: 3 k-steps x 4 n-tiles = 12 WMMAs.
__global__ __launch_bounds__(256) void mdmlp_patch_wmma_f32(
    const float* __restrict__ x,      // [16,64,56,56]
    const float* __restrict__ Wm,     // [9,64] row-major
    const float* __restrict__ bias,   // [64]
    float* __restrict__ out)          // [16,64,3136,64]
{
    const int lane = threadIdx.x & 31;
    const int waveInBlock = threadIdx.x >> 5;

    const long long tile = (long long)blockIdx.x * 8 + waveInBlock;
    const long long m0   = tile * 16;            // first GEMM row of this tile

    // decode (n, c, p0); tiles never straddle (n,c) since P_TOT % 16 == 0
    const long long nc = m0 / P_TOT;
    const int p0 = (int)(m0 - nc * P_TOT);
    const int c  = (int)(nc & (C_CH - 1));
    const int n  = (int)(nc >> 6);

    // A-matrix 16x4 f32 layout: lanes 0-15 hold K=0,1 ; lanes 16-31 hold K=2,3
    const int mrow = lane & 15;                  // M within tile (also B column id)
    const int koff = (lane >> 4) << 1;           // 0 or 2

    // ---- gather 3x3 zero-padded window for row p0+mrow (stays hot in L2) ----
    const int p  = p0 + mrow;
    const int oh = p / WW;
    const int ow = p - oh * WW;
    const float* xc = x + ((long long)n * C_CH + c) * (HH * WW);

    float patch[12];
#pragma unroll
    for (int i = 0; i < 3; ++i) {
#pragma unroll
        for (int j = 0; j < 3; ++j) {
            const int ih = oh + i - 1;
            const int iw = ow + j - 1;
            float v = 0.0f;
            if (ih >= 0 && ih < HH && iw >= 0 && iw < WW)
                v = xc[ih * WW + iw];
            patch[i * 3 + j] = v;
        }
    }
    patch[9] = 0.0f; patch[10] = 0.0f; patch[11] = 0.0f;   // K padded 9 -> 12

    // ---- B-matrix 4x16 f32 regs: V0/V1 = rows koff / koff+1, col = nt*16+mrow
    v2f bmat[4][3];
#pragma unroll
    for (int nt = 0; nt < 4; ++nt) {
        const int col = nt * 16 + mrow;
#pragma unroll
        for (int kk = 0; kk < 3; ++kk) {
            const int k0 = 4 * kk + koff;
            const int k1 = k0 + 1;
            v2f b;
            b.x = (k0 < 9) ? Wm[k0 * 64 + col] : 0.0f;
            b.y = (k1 < 9) ? Wm[k1 * 64 + col] : 0.0f;
            bmat[nt][kk] = b;
        }
    }

    // ---- 12x V_WMMA_F32_16X16X4_F32 (EXEC is all-1s: exact grid, no guards)
    v8f acc[4] = {};
#pragma unroll
    for (int kk = 0; kk < 3; ++kk) {
        v2f a;
        a.x = patch[4 * kk + koff];
        a.y = patch[4 * kk + koff + 1];
#pragma unroll
        for (int nt = 0; nt < 4; ++nt) {
            acc[nt] = __builtin_amdgcn_wmma_f32_16x16x4_f32(
                /*neg_a=*/false, a, /*neg_b=*/false, bmat[nt][kk],
                /*c_mod=*/(short)0, acc[nt], /*reuse_a=*/false, /*reuse_b=*/false);
        }
    }

    // ---- bias + non-temporal streaming stores (822 MB write-once: bypass L2
    //      retention; 64B-contiguous per half-wave, 256B per row across nt)
    const int mhi = (lane >> 4) << 3;            // 0 or 8
#pragma unroll
    for (int nt = 0; nt < 4; ++nt) {
        const int col = nt * 16 + mrow;
        const float bv = bias[col];
        float* op = out + m0 * 64 + col;
#pragma unroll
        for (int r = 0; r < 8; ++r) {
            const long long row = (long long)(r + mhi);
            __builtin_nontemporal_store(acc[nt][r] + bv, op + row * 64);
        }
    }
}

extern "C" void kernel_launch(void* const* d_in, const int* in_sizes, int n_in,
                              void* d_out, int out_size, void* d_ws, size_t ws_size,
                              hipStream_t stream) {
    const float* x  = (const float*)d_in[0];   // [16,64,56,56] f32
    const float* Wm = (const float*)d_in[1];   // [9,64] f32
    const float* bv = (const float*)d_in[2];   // [64] f32
    float* out = (float*)d_out;                // [16,64,3136,64] f32

    const long long M = (long long)in_sizes[0];        // N*C*H*W = 3,211,264 rows
    const long long tiles = M / 16;                    // 200,704 waves
    const int blocks = (int)(tiles / 8);               // 8 waves (256 thr) / block

    mdmlp_patch_wmma_f32<<<blocks, 256, 0, stream>>>(x, Wm, bv, out);
}